// KVCache_79276506350246
// MI455X (gfx1250) — compile-verified
//
#include <hip/hip_runtime.h>

// KV-cache append: out[1,8,65552,64] = concat(k_cache[1,8,65536,64], x[1,8,16,64], dim=-2)
// Pure streaming copy, ~268 MB of traffic -> HBM-bound (~11.5 us at 23.3 TB/s).
// Bulk copy uses the CDNA5 async data-mover path:
//   global_load_async_to_lds_b128  -> LDS -> global_store_async_from_lds_b128
// tracked with ASYNCcnt (s_wait_asynccnt), 16 B per lane per instruction.

#define CACHE_F4_PER_HEAD (65536u * 64u / 4u)  // 1,048,576 = 2^20 float4 per head (input)
#define OUT_F4_PER_HEAD   (65552u * 64u / 4u)  // 1,048,832 float4 per head (output)
#define X_F4_PER_HEAD     (16u * 64u / 4u)     // 256 float4 per head (appended tokens)

// Bulk cache copy. Grid: 8192 blocks x 256 threads, 4 slots of 16B per thread:
// 8192 * 256 * 4 = 8,388,608 float4 = entire cache. 1024 float4 per block, and
// 1024 | 2^20, so a block never straddles a head boundary -> contiguous in src & dst.
__global__ void kv_cache_copy_async(const float* __restrict__ kc, float* __restrict__ out) {
    __shared__ char lbuf[16384];  // 256 threads * 4 slots * 16 B

    unsigned tid  = threadIdx.x;
    unsigned i0   = blockIdx.x * 1024u + tid;            // slot-0 float4 index (flat cache)
    unsigned head = i0 >> 20;                            // uniform within block
    unsigned rem  = i0 & (CACHE_F4_PER_HEAD - 1u);

    unsigned long long src =
        (unsigned long long)(size_t)kc + (unsigned long long)i0 * 16ull;
    unsigned long long dst =
        (unsigned long long)(size_t)out +
        ((unsigned long long)head * OUT_F4_PER_HEAD + rem) * 16ull;

    // Flat LDS aperture keeps the LDS byte offset in the low 32 bits.
    unsigned lds = (unsigned)(size_t)(void*)(lbuf + tid * 16u);

    // ISA: INST_OFFSET is added to BOTH the global and LDS addresses, so the
    // same (lds, src/dst) VGPRs cover all 4 slots via offset immediates
    // (slot stride = 256 lanes * 16 B = 4096 B in LDS and in memory).
    asm volatile(
        "global_load_async_to_lds_b128 %0, %1, off\n\t"
        "global_load_async_to_lds_b128 %0, %1, off offset:4096\n\t"
        "global_load_async_to_lds_b128 %0, %1, off offset:8192\n\t"
        "global_load_async_to_lds_b128 %0, %1, off offset:12288\n\t"
        "s_wait_asynccnt 0x0\n\t"
        "global_store_async_from_lds_b128 %2, %0, off\n\t"
        "global_store_async_from_lds_b128 %2, %0, off offset:4096\n\t"
        "global_store_async_from_lds_b128 %2, %0, off offset:8192\n\t"
        "global_store_async_from_lds_b128 %2, %0, off offset:12288\n\t"
        :
        : "v"(lds), "v"(src), "v"(dst)
        : "memory");
    // S_ENDPGM performs an implicit wait-idle, covering outstanding async stores.
}

// Append x: 8 heads * 256 float4 = 2048 float4 total (tiny). Plain b128 copy.
__global__ void kv_x_append(const float4* __restrict__ x, float4* __restrict__ out) {
    unsigned t    = blockIdx.x * blockDim.x + threadIdx.x;  // 0..2047
    unsigned head = t >> 8;
    unsigned rem  = t & 255u;
    out[(size_t)head * OUT_F4_PER_HEAD + CACHE_F4_PER_HEAD + rem] = x[t];
}

extern "C" void kernel_launch(void* const* d_in, const int* in_sizes, int n_in,
                              void* d_out, int out_size, void* d_ws, size_t ws_size,
                              hipStream_t stream) {
    const float* x  = (const float*)d_in[0];  // [1,8,16,64] fp32
    const float* kc = (const float*)d_in[1];  // [1,8,65536,64] fp32
    float* out = (float*)d_out;               // [1,8,65552,64] fp32

    kv_cache_copy_async<<<8192, 256, 0, stream>>>(kc, out);
    kv_x_append<<<8, 256, 0, stream>>>((const float4*)x, (float4*)out);
}